// DataConsistencyMoDL_13451837571301
// MI455X (gfx1250) — compile-verified
//
#include <hip/hip_runtime.h>

typedef float v2f __attribute__((ext_vector_type(2)));
typedef float v8f __attribute__((ext_vector_type(8)));
typedef unsigned int v4u __attribute__((ext_vector_type(4)));
typedef unsigned int v8u __attribute__((ext_vector_type(8)));

#define FN    2048
#define ROWS  8
#define RSTR  2052                   /* float2 per LDS row: 2048 + 4 pad */
#define LDS_BYTES (ROWS * RSTR * 8)  /* 131328 bytes */
#define LAM   0.05f
#define NBE   4096                   /* elementwise grid */

static __device__ __forceinline__ float2 cxadd(float2 a, float2 b){ return make_float2(a.x+b.x, a.y+b.y); }
static __device__ __forceinline__ float2 cxsub(float2 a, float2 b){ return make_float2(a.x-b.x, a.y-b.y); }
static __device__ __forceinline__ float2 cxmul(float2 a, float2 b){ return make_float2(a.x*b.x-a.y*b.y, a.x*b.y+a.y*b.x); }
static __device__ __forceinline__ float2 cxrot(float2 a, float s){ return make_float2(-s*a.y, s*a.x); } // * (s*i)

static __device__ __forceinline__ v8f wmma4(v2f a, v2f b, v8f c) {
  // D = A(16x4,f32) * B(4x16,f32) + C(16x16,f32)
  return __builtin_amdgcn_wmma_f32_16x16x4_f32(false, a, false, b, (short)0, c, false, false);
}

// 8-point DFT, in place; sign s = -1 forward, +1 inverse (unnormalized)
static __device__ __forceinline__ void dft8(float2 a[8], float s) {
  float2 t0=cxadd(a[0],a[4]), t1=cxsub(a[0],a[4]);
  float2 t2=cxadd(a[2],a[6]), t3=cxsub(a[2],a[6]);
  float2 t4=cxadd(a[1],a[5]), t5=cxsub(a[1],a[5]);
  float2 t6=cxadd(a[3],a[7]), t7=cxsub(a[3],a[7]);
  float2 r3=cxrot(t3,s), r7=cxrot(t7,s);
  float2 e0=cxadd(t0,t2), e2=cxsub(t0,t2);
  float2 e1=cxadd(t1,r3), e3=cxsub(t1,r3);
  float2 o0=cxadd(t4,t6), o2=cxsub(t4,t6);
  float2 o1=cxadd(t5,r7), o3=cxsub(t5,r7);
  const float C = 0.70710678118654752f;
  float2 w1o1 = make_float2(C*(o1.x - s*o1.y), C*(o1.y + s*o1.x));   // * C(1+si)
  float2 ro2  = cxrot(o2, s);
  float2 w3o3 = make_float2(C*(s*o3.y - o3.x), C*(s*o3.x - o3.y));   // * C(-1+si)
  a[0]=cxadd(e0,o0);   a[4]=cxsub(e0,o0);
  a[1]=cxadd(e1,w1o1); a[5]=cxsub(e1,w1o1);
  a[2]=cxadd(e2,ro2);  a[6]=cxsub(e2,ro2);
  a[3]=cxadd(e3,w3o3); a[7]=cxsub(e3,w3o3);
}

// One pass P: for 8 rows r of src, compute 2048-pt DFT of src[row][:], write
// dst[k][row] (transposed). P(P(x)) == fft2(x). Optional W multiply + scale at store.
__global__ __launch_bounds__(128) void fft_pass_kernel(const float2* __restrict__ src,
                                                       float2* __restrict__ dst,
                                                       const float* __restrict__ Wdc,
                                                       float sgn, float scale) {
  extern __shared__ float2 tile[];           // [ROWS][RSTR]
  const int tid  = threadIdx.x;
  const int lane = tid & 31;
  const int wave = tid >> 5;                 // 0..3
  const int hi   = lane >> 4;                // 0/1
  const int l16  = lane & 15;
  const int row0 = blockIdx.x * ROWS;

  // ---- Tensor Data Mover: DMA the 8 source rows into (padded) LDS ----
  if (wave == 0) {
#pragma unroll
    for (int r = 0; r < ROWS; ++r) {
      unsigned long long ga = (unsigned long long)(uintptr_t)(src + (size_t)(row0 + r) * FN);
      unsigned int la = (unsigned int)(uintptr_t)(&tile[r * RSTR]);
      v4u g0;
      g0[0] = 1u;                                        // count=1
      g0[1] = la;                                        // lds_addr
      g0[2] = (unsigned)(ga & 0xffffffffull);            // global_addr[31:0]
      g0[3] = (unsigned)((ga >> 32) & 0x1ffffffull) | (2u << 30); // addr[56:32] | type=2
      v8u g1;
      g1[0] = (3u << 16);                                // data_size=8B
      g1[1] = ((unsigned)FN & 0xffffu) << 16;            // tensor_dim0 lo16 = 2048
      g1[2] = 1u << 16;                                  // tensor_dim0 hi=0 | tensor_dim1=1
      g1[3] = ((unsigned)FN) << 16;                      // tile_dim0 = 2048
      g1[4] = 1u;                                        // tile_dim1=1, tile_dim2=0
      g1[5] = (unsigned)FN;                              // tensor_dim0_stride = 2048
      g1[6] = 0u; g1[7] = 0u;
      asm volatile("tensor_load_to_lds %0, %1" :: "s"(g0), "s"(g1) : "memory");
    }
    __builtin_amdgcn_s_wait_tensorcnt(0);
  }
  __syncthreads();

  // Hoisted DFT-16 matrix fragments (A operand, M=l16, K=4c+v+2*hi)
  float Ar[4][2], Ai[4][2];
#pragma unroll
  for (int c = 0; c < 4; ++c)
#pragma unroll
    for (int v = 0; v < 2; ++v) {
      int K = 4*c + v + 2*hi;
      float ang = 0.39269908169872414f * (float)((l16 * K) & 15);   // 2pi/16
      float si, co; __sincosf(ang, &si, &co);
      Ar[c][v] = co; Ai[c][v] = sgn * si;
    }

  for (int r = 0; r < ROWS; ++r) {
    float2* row = &tile[r * RSTR];

    // ---- Stage A: 256 DFT8s + twiddle E(m1*j2/128), layout -> [m1][n1*8+j2]
    float2 va[2][8];
#pragma unroll
    for (int pp = 0; pp < 2; ++pp) {
      int p = tid + pp * 128, n1 = p & 15, m1 = p >> 4;
#pragma unroll
      for (int m2 = 0; m2 < 8; ++m2) va[pp][m2] = row[n1 + 16*m1 + 256*m2];
    }
    __syncthreads();
#pragma unroll
    for (int pp = 0; pp < 2; ++pp) {
      int p = tid + pp * 128, n1 = p & 15, m1 = p >> 4;
      dft8(va[pp], sgn);
      int base = m1*128 + n1*8;
#pragma unroll
      for (int j2 = 0; j2 < 8; ++j2) {
        float ang = 0.04908738521234052f * (float)(m1 * j2);        // 2pi/128
        float si, co; __sincosf(ang, &si, &co); si *= sgn;
        row[base + j2] = cxmul(va[pp][j2], make_float2(co, si));
      }
    }
    __syncthreads();

    // ---- Stage B: DFT16 over m1 (WMMA) + twiddle E(n1*k2/2048) -> [n1][k2]
    float2 bf[2][4][2];
#pragma unroll
    for (int tt = 0; tt < 2; ++tt) {
      int col = (wave*2 + tt) * 16 + l16;
#pragma unroll
      for (int c = 0; c < 4; ++c)
#pragma unroll
        for (int v = 0; v < 2; ++v) bf[tt][c][v] = row[(4*c + v + 2*hi)*128 + col];
    }
    __syncthreads();
#pragma unroll
    for (int tt = 0; tt < 2; ++tt) {
      v8f accR = {0,0,0,0,0,0,0,0}, accI = {0,0,0,0,0,0,0,0};
#pragma unroll
      for (int c = 0; c < 4; ++c) {
        v2f ar = {Ar[c][0], Ar[c][1]}, ai = {Ai[c][0], Ai[c][1]};
        v2f br = {bf[tt][c][0].x, bf[tt][c][1].x};
        v2f bi = {bf[tt][c][0].y, bf[tt][c][1].y};
        v2f bn = -bi;
        accR = wmma4(ar, br, accR);  accR = wmma4(ai, bn, accR);
        accI = wmma4(ar, bi, accI);  accI = wmma4(ai, br, accI);
      }
      int col = (wave*2 + tt) * 16 + l16, n1 = col >> 3, j2 = col & 7;
#pragma unroll
      for (int q = 0; q < 8; ++q) {
        int j1 = q + 8*hi, k2 = j2 + 8*j1;
        float ang = 0.003067961575771282f * (float)(n1 * k2);       // 2pi/2048
        float si, co; __sincosf(ang, &si, &co); si *= sgn;
        row[n1*128 + k2] = cxmul(make_float2(accR[q], accI[q]), make_float2(co, si));
      }
    }
    __syncthreads();

    // ---- Stage C: DFT16 over n1 (WMMA) -> X at k2 + 128*k1
#pragma unroll
    for (int tt = 0; tt < 2; ++tt) {
      int col = (wave*2 + tt) * 16 + l16;
#pragma unroll
      for (int c = 0; c < 4; ++c)
#pragma unroll
        for (int v = 0; v < 2; ++v) bf[tt][c][v] = row[(4*c + v + 2*hi)*128 + col];
    }
    __syncthreads();
#pragma unroll
    for (int tt = 0; tt < 2; ++tt) {
      v8f accR = {0,0,0,0,0,0,0,0}, accI = {0,0,0,0,0,0,0,0};
#pragma unroll
      for (int c = 0; c < 4; ++c) {
        v2f ar = {Ar[c][0], Ar[c][1]}, ai = {Ai[c][0], Ai[c][1]};
        v2f br = {bf[tt][c][0].x, bf[tt][c][1].x};
        v2f bi = {bf[tt][c][0].y, bf[tt][c][1].y};
        v2f bn = -bi;
        accR = wmma4(ar, br, accR);  accR = wmma4(ai, bn, accR);
        accI = wmma4(ar, bi, accI);  accI = wmma4(ai, br, accI);
      }
      int k2 = (wave*2 + tt) * 16 + l16;
#pragma unroll
      for (int q = 0; q < 8; ++q) {
        int k1 = q + 8*hi;
        row[k2 + 128*k1] = make_float2(accR[q], accI[q]);
      }
    }
    __syncthreads();
  }

  // ---- transposed, coalesced store: dst[k*FN + row0 + rr] (optional W, scale)
  for (int i = tid; i < ROWS * FN; i += 128) {
    int k = i >> 3, rr = i & 7;
    float2 v = tile[rr * RSTR + k];
    size_t j = (size_t)k * FN + (size_t)(row0 + rr);
    float sc = scale;
    if (Wdc) sc *= Wdc[j];
    dst[j] = make_float2(v.x * sc, v.y * sc);
  }
}

// ------------------------- utility kernels -------------------------
__global__ __launch_bounds__(256) void zero_f32(float* p, int n) {
  for (int i = blockIdx.x*blockDim.x + threadIdx.x; i < n; i += gridDim.x*blockDim.x) p[i] = 0.f;
}
__global__ __launch_bounds__(256) void hist_k(const int* __restrict__ idx, float* __restrict__ W, int n) {
  int i = blockIdx.x*blockDim.x + threadIdx.x;
  if (i < n) atomicAdd(&W[idx[i]], 1.0f);
}
__global__ __launch_bounds__(256) void scatter_k(const int* __restrict__ idx, const float* __restrict__ yr,
                                                 const float* __restrict__ yi, float2* __restrict__ g, int n) {
  int i = blockIdx.x*blockDim.x + threadIdx.x;
  if (i < n) { int j = idx[i]; atomicAdd(&g[j].x, yr[i]); atomicAdd(&g[j].y, yi[i]); }
}

static __device__ __forceinline__ void block_reduce_store(float acc, float* partial) {
  __shared__ float sred[256];
  sred[threadIdx.x] = acc; __syncthreads();
  for (int s = 128; s > 0; s >>= 1) { if ((int)threadIdx.x < s) sred[threadIdx.x] += sred[threadIdx.x+s]; __syncthreads(); }
  if (threadIdx.x == 0) partial[blockIdx.x] = sred[0];
}

__global__ __launch_bounds__(256) void init_cg_k(const float2* __restrict__ binv, const float* __restrict__ zkr,
                                                 const float* __restrict__ zki, float2* __restrict__ x,
                                                 float2* __restrict__ r, float2* __restrict__ p,
                                                 float* __restrict__ partial, int n) {
  float acc = 0.f;
  for (int i = blockIdx.x*blockDim.x + threadIdx.x; i < n; i += gridDim.x*blockDim.x) {
    float2 bb = make_float2(binv[i].x + LAM*zkr[i], binv[i].y + LAM*zki[i]);
    r[i] = bb; p[i] = bb; x[i] = make_float2(0.f, 0.f);
    acc += bb.x*bb.x + bb.y*bb.y;
  }
  block_reduce_store(acc, partial);
}
__global__ __launch_bounds__(256) void pAp_k(float2* __restrict__ Ap, const float2* __restrict__ p,
                                             float* __restrict__ partial, int n) {
  float acc = 0.f;
  for (int i = blockIdx.x*blockDim.x + threadIdx.x; i < n; i += gridDim.x*blockDim.x) {
    float2 pv = p[i], ap = Ap[i];
    ap.x += LAM*pv.x; ap.y += LAM*pv.y; Ap[i] = ap;
    acc += pv.x*ap.x + pv.y*ap.y;
  }
  block_reduce_store(acc, partial);
}
__global__ __launch_bounds__(256) void update_xr_k(float2* __restrict__ x, float2* __restrict__ r,
                                                   const float2* __restrict__ p, const float2* __restrict__ Ap,
                                                   const float* __restrict__ sc, float* __restrict__ partial, int n) {
  float a = sc[1], acc = 0.f;
  for (int i = blockIdx.x*blockDim.x + threadIdx.x; i < n; i += gridDim.x*blockDim.x) {
    float2 xv = x[i], pv = p[i], rv = r[i], ap = Ap[i];
    xv.x += a*pv.x; xv.y += a*pv.y; x[i] = xv;
    rv.x -= a*ap.x; rv.y -= a*ap.y; r[i] = rv;
    acc += rv.x*rv.x + rv.y*rv.y;
  }
  block_reduce_store(acc, partial);
}
__global__ __launch_bounds__(256) void update_p_k(float2* __restrict__ p, const float2* __restrict__ r,
                                                  const float* __restrict__ sc, int n) {
  float b = sc[2];
  for (int i = blockIdx.x*blockDim.x + threadIdx.x; i < n; i += gridDim.x*blockDim.x) {
    float2 rv = r[i], pv = p[i];
    p[i] = make_float2(rv.x + b*pv.x, rv.y + b*pv.y);
  }
}
__global__ __launch_bounds__(256) void writeout_k(const float2* __restrict__ x, float* __restrict__ out, int n) {
  for (int i = blockIdx.x*blockDim.x + threadIdx.x; i < n; i += gridDim.x*blockDim.x) {
    out[i] = x[i].x; out[n + i] = x[i].y;
  }
}
// mode 0: rs_old = total ; mode 1: alpha = rs_old/total ; mode 2: beta = total/rs_old, rs_old = total
__global__ __launch_bounds__(256) void finalize_k(const float* __restrict__ partial, int n,
                                                  float* __restrict__ sc, int mode) {
  float acc = 0.f;
  for (int i = threadIdx.x; i < n; i += 256) acc += partial[i];
  __shared__ float sred[256];
  sred[threadIdx.x] = acc; __syncthreads();
  for (int s = 128; s > 0; s >>= 1) { if ((int)threadIdx.x < s) sred[threadIdx.x] += sred[threadIdx.x+s]; __syncthreads(); }
  if (threadIdx.x == 0) {
    float total = sred[0];
    if (mode == 0)      { sc[0] = total; }
    else if (mode == 1) { sc[1] = sc[0] / total; }
    else                { sc[2] = total / sc[0]; sc[0] = total; }
  }
}

extern "C" void kernel_launch(void* const* d_in, const int* in_sizes, int n_in,
                              void* d_out, int out_size, void* d_ws, size_t ws_size,
                              hipStream_t stream) {
  const float* zkr = (const float*)d_in[0];
  const float* zki = (const float*)d_in[1];
  const float* yr  = (const float*)d_in[2];
  const float* yi  = (const float*)d_in[3];
  const int*   idx = (const int*)d_in[4];
  const int nk = in_sizes[2];
  const int N2 = FN * FN;

  char* ws = (char*)d_ws;
  float*  Wb = (float*)ws;  ws += (size_t)N2 * 4;
  float2* t0 = (float2*)ws; ws += (size_t)N2 * 8;
  float2* t1 = (float2*)ws; ws += (size_t)N2 * 8;
  float2* xv = (float2*)ws; ws += (size_t)N2 * 8;
  float2* rv = (float2*)ws; ws += (size_t)N2 * 8;
  float2* pv = (float2*)ws; ws += (size_t)N2 * 8;
  float2* Ap = (float2*)ws; ws += (size_t)N2 * 8;
  float* partial = (float*)ws; ws += (size_t)NBE * 4;
  float* sc = (float*)ws;
  float* out = (float*)d_out;

  hipFuncSetAttribute((const void*)fft_pass_kernel,
                      hipFuncAttributeMaxDynamicSharedMemorySize, LDS_BYTES);

  const dim3 fg(FN / ROWS), fb(128), eg(NBE), eb(256);
  const int hb = (nk + 255) / 256;

  // W = histogram(idx); g = scatter(y)
  zero_f32<<<8192, 256, 0, stream>>>(Wb, N2);
  zero_f32<<<8192, 256, 0, stream>>>((float*)t1, 2 * N2);
  hist_k<<<hb, 256, 0, stream>>>(idx, Wb, nk);
  scatter_k<<<hb, 256, 0, stream>>>(idx, yr, yi, t1, nk);

  // binv = ifft2_ortho(g)  (unnormalized inverse, total scale 1/2048)
  fft_pass_kernel<<<fg, fb, LDS_BYTES, stream>>>(t1, t0, nullptr, 1.f, 1.f);
  fft_pass_kernel<<<fg, fb, LDS_BYTES, stream>>>(t0, Ap, nullptr, 1.f, 1.f / 2048.f);
  // b = binv + lam*zk ; x=0 ; r=p=b ; rs_old = |b|^2
  init_cg_k<<<eg, eb, 0, stream>>>(Ap, zkr, zki, xv, rv, pv, partial, N2);
  finalize_k<<<1, 256, 0, stream>>>(partial, NBE, sc, 0);

  for (int it = 0; it < 5; ++it) {
    // Ap = ifft2(W .* fft2(p)) / 2048^2   (then += lam*p inside pAp_k)
    fft_pass_kernel<<<fg, fb, LDS_BYTES, stream>>>(pv, t0, nullptr, -1.f, 1.f);
    fft_pass_kernel<<<fg, fb, LDS_BYTES, stream>>>(t0, t1, Wb, -1.f, 1.f);
    fft_pass_kernel<<<fg, fb, LDS_BYTES, stream>>>(t1, t0, nullptr, 1.f, 1.f);
    fft_pass_kernel<<<fg, fb, LDS_BYTES, stream>>>(t0, Ap, nullptr, 1.f, 1.f / (2048.f * 2048.f));
    pAp_k<<<eg, eb, 0, stream>>>(Ap, pv, partial, N2);
    finalize_k<<<1, 256, 0, stream>>>(partial, NBE, sc, 1);            // alpha
    update_xr_k<<<eg, eb, 0, stream>>>(xv, rv, pv, Ap, sc, partial, N2);
    finalize_k<<<1, 256, 0, stream>>>(partial, NBE, sc, 2);            // beta, rs_old
    update_p_k<<<eg, eb, 0, stream>>>(pv, rv, sc, N2);
  }

  writeout_k<<<eg, eb, 0, stream>>>(xv, out, N2);
}